// LSNN_11347303596361
// MI455X (gfx1250) — compile-verified
//
#include <hip/hip_runtime.h>

typedef float v2f __attribute__((ext_vector_type(2)));
typedef float v8f __attribute__((ext_vector_type(8)));

static constexpr int kB = 32768;   // batch
static constexpr int kT = 50;      // timesteps
static constexpr int kH = 32;      // hidden
static constexpr int kLdsStride = 34;  // floats per h-row in LDS (bank-conflict free)

__device__ __forceinline__ float sigmoid_f(float x) {
  return __builtin_amdgcn_rcpf(1.0f + __expf(-x));
}
__device__ __forceinline__ float tanh_f(float x) {
  // tanh(x) = 1 - 2/(exp(2x)+1); saturates correctly for large |x|
  return 1.0f - 2.0f * __builtin_amdgcn_rcpf(__expf(2.0f * x) + 1.0f);
}

// One wave handles 16 batch rows for all 50 timesteps.
// Per timestep:  gates^T[128x16] = W_aug[128x36] x [h; x0; x1; 1]^T[36x16]
// via 8 tiles x 9 ksteps of V_WMMA_F32_16X16X4_F32 (full fp32 precision).
__global__ __launch_bounds__(128, 1) void lstm_wmma_f32(
    const float* __restrict__ x,      // [B, T, 2]
    const float* __restrict__ W_ih,   // [128, 2]
    const float* __restrict__ W_hh,   // [128, 32]
    const float* __restrict__ b_ih,   // [128]
    const float* __restrict__ b_hh,   // [128]
    const float* __restrict__ W_out,  // [1, 32]
    const float* __restrict__ b_out,  // [1]
    const float* __restrict__ h0,     // [B, 32]
    const float* __restrict__ c0,     // [B, 32]
    float* __restrict__ out)          // [B, T]
{
  __shared__ __align__(16) float lds_h[4][16 * kLdsStride];

  const int lane = threadIdx.x & 31;
  const int wave = threadIdx.x >> 5;
  const int n    = lane & 15;   // batch column within tile / A-matrix row
  const int hi   = lane >> 4;   // lane half (selects K/M sub-block)
  const int batch_base = (blockIdx.x * 4 + wave) * 16;
  if (batch_base >= kB) return;
  const int bg = batch_base + n;
  float* myLds = &lds_h[wave][0];

  // ---- Preload A fragments of W_aug[128 x 36] ----------------------------
  // A-layout (16x4 f32): lane holds row M = 16*tile + n, k = 4*ks + 2*hi + v
  // k<32 -> W_hh ; k==32,33 -> W_ih cols ; k==34 -> b_ih+b_hh ; k==35 -> 0
  v2f a[8][9];
#pragma unroll
  for (int t8 = 0; t8 < 8; ++t8) {
    const int g = 16 * t8 + n;
#pragma unroll
    for (int ks = 0; ks < 8; ++ks)
      a[t8][ks] = *(const v2f*)(W_hh + g * kH + 4 * ks + 2 * hi);
    if (hi == 0) {
      a[t8][8] = *(const v2f*)(W_ih + g * 2);            // {W_ih[g][0], W_ih[g][1]}
    } else {
      a[t8][8] = v2f{b_ih[g] + b_hh[g], 0.0f};           // {bias, pad}
    }
  }

  // ---- Preload output-projection weights for this lane's k-set -----------
  float wout[16];
#pragma unroll
  for (int tp = 0; tp < 2; ++tp)
#pragma unroll
    for (int r = 0; r < 8; ++r)
      wout[8 * tp + r] = W_out[16 * tp + 8 * hi + r];
  const float bo = b_out[0];

  // ---- Initial state ------------------------------------------------------
  // B-layout fragment of h: lane holds column n, k = 4*ks + 2*hi + v
  v2f bfrag[9];
#pragma unroll
  for (int ks = 0; ks < 8; ++ks)
    bfrag[ks] = *(const v2f*)(h0 + bg * kH + 4 * ks + 2 * hi);
  // c in D-layout: lane holds (batch n, k = 16*tp + 8*hi + r)
  float c[16];
#pragma unroll
  for (int tp = 0; tp < 2; ++tp)
#pragma unroll
    for (int r = 0; r < 8; ++r)
      c[8 * tp + r] = c0[bg * kH + 16 * tp + 8 * hi + r];

  const v2f one_zero = v2f{1.0f, 0.0f};

  for (int t = 0; t < kT; ++t) {
    // B rows 32..35: {x0, x1} for lane-half 0, {1, 0} for lane-half 1
    const v2f xv = *(const v2f*)(x + (bg * kT + t) * 2);
    bfrag[8] = hi ? one_zero : xv;

    // ---- gates^T = W_aug x [h;x;1]^T : 8 tiles x 9 ksteps of fp32 WMMA ---
    v8f acc[8];
#pragma unroll
    for (int tile = 0; tile < 8; ++tile) {
      v8f d = {};
#pragma unroll
      for (int ks = 0; ks < 9; ++ks)
        d = __builtin_amdgcn_wmma_f32_16x16x4_f32(
            false, a[tile][ks], false, bfrag[ks], (short)0, d, false, false);
      acc[tile] = d;
    }

    // ---- lane-local LSTM cell update (gate order i,f,g,o) ----------------
    // acc tile 0..1 = i, 2..3 = f, 4..5 = g, 6..7 = o ; element r, half hi
    float hd[16];
    float partial = 0.0f;
#pragma unroll
    for (int tp = 0; tp < 2; ++tp) {
#pragma unroll
      for (int r = 0; r < 8; ++r) {
        const float iv = sigmoid_f(acc[0 + tp][r]);
        const float fv = sigmoid_f(acc[2 + tp][r]);
        const float gv = tanh_f(acc[4 + tp][r]);
        const float ov = sigmoid_f(acc[6 + tp][r]);
        const float cv = fv * c[8 * tp + r] + iv * gv;
        c[8 * tp + r] = cv;
        const float hv = ov * tanh_f(cv);
        hd[8 * tp + r] = hv;
        partial += wout[8 * tp + r] * hv;
      }
    }

    // ---- D-layout -> B-layout reshape of h through per-wave LDS slice ----
    // Same-wave DS ops execute in order (ISA 7.3); fences stop compiler
    // reordering across the aliasing store/load groups.
    asm volatile("" ::: "memory");
#pragma unroll
    for (int tp = 0; tp < 2; ++tp)
#pragma unroll
      for (int j = 0; j < 4; ++j)
        *(v2f*)(myLds + n * kLdsStride + 16 * tp + 8 * hi + 2 * j) =
            v2f{hd[8 * tp + 2 * j], hd[8 * tp + 2 * j + 1]};
    asm volatile("" ::: "memory");
#pragma unroll
    for (int ks = 0; ks < 8; ++ks)
      bfrag[ks] = *(const v2f*)(myLds + n * kLdsStride + 4 * ks + 2 * hi);
    asm volatile("" ::: "memory");

    // ---- output projection: reduce this lane's 16 k's with partner lane --
    const float other = __shfl_xor(partial, 16, 32);
    if (lane < 16) out[bg * kT + t] = partial + other + bo;
  }
}

extern "C" void kernel_launch(void* const* d_in, const int* in_sizes, int n_in,
                              void* d_out, int out_size, void* d_ws,
                              size_t ws_size, hipStream_t stream) {
  const float* x     = (const float*)d_in[0];
  const float* W_ih  = (const float*)d_in[1];
  const float* W_hh  = (const float*)d_in[2];
  const float* b_ih  = (const float*)d_in[3];
  const float* b_hh  = (const float*)d_in[4];
  const float* W_out = (const float*)d_in[5];
  const float* b_out = (const float*)d_in[6];
  const float* h0    = (const float*)d_in[7];
  const float* c0    = (const float*)d_in[8];
  float* out = (float*)d_out;

  dim3 grid(kB / 64);   // 512 blocks, 4 waves each, 16 batch rows per wave
  dim3 block(128);
  lstm_wmma_f32<<<grid, block, 0, stream>>>(x, W_ih, W_hh, b_ih, b_hh, W_out,
                                            b_out, h0, c0, out);
}